// Custom_Loss_3_42545946034228
// MI455X (gfx1250) — compile-verified
//
#include <hip/hip_runtime.h>

// L1-loss mean reduction, MI455X (gfx1250, wave32).
// Pass 1: stream |a-b| with NT b128 loads, per-lane f32 accumulate,
//         wave reduce via V_WMMA_F32_16X16X4_F32 (B = ones trick),
//         cross-wave via LDS -> one partial per block.
// Pass 2: one block reduces all partials, scales by 1/N_VEH.

typedef __attribute__((ext_vector_type(2))) float v2f;
typedef __attribute__((ext_vector_type(4))) float v4f;
typedef __attribute__((ext_vector_type(8))) float v8f;

// Full 32-lane sum of `acc`, returned in every lane.
// A (16x4 f32): lane L<16 holds {A[L][0]=acc, A[L][1]=0};
//               lane L>=16 holds {A[L-16][2]=acc, A[L-16][3]=0}.
// B (4x16) = all ones  =>  D[m][n] = acc_m + acc_{m+16} (same for all n).
// Lanes 0-15 see rows m=0..7 in their 8 D VGPRs, lanes 16-31 see m=8..15,
// so one xor-16 shuffle completes the reduction at full f32 precision.
__device__ __forceinline__ float wave_sum_wmma(float acc) {
    v2f a;  a[0] = acc;  a[1] = 0.0f;
    v2f b;  b[0] = 1.0f; b[1] = 1.0f;
    v8f c = {};
    v8f d = __builtin_amdgcn_wmma_f32_16x16x4_f32(
        /*neg_a=*/false, a, /*neg_b=*/false, b,
        /*c_mod=*/(short)0, c, /*reuse_a=*/false, /*reuse_b=*/false);
    float s = ((d[0] + d[1]) + (d[2] + d[3])) + ((d[4] + d[5]) + (d[6] + d[7]));
    s += __shfl_xor(s, 16, 32);
    return s;
}

__global__ void __launch_bounds__(256)
l1_partial_kernel(const float* __restrict__ a, const float* __restrict__ b,
                  float* __restrict__ partials, long long n, int iters) {
    const int tid      = blockIdx.x * blockDim.x + threadIdx.x;
    const int nthreads = gridDim.x * blockDim.x;

    const v4f* __restrict__ av = (const v4f*)a;
    const v4f* __restrict__ bv = (const v4f*)b;

    // 4 independent accumulators (ILP + better summation tree).
    float s0 = 0.0f, s1 = 0.0f, s2 = 0.0f, s3 = 0.0f;

    // Uniform main loop: every lane runs exactly `iters` b128+b128 loads.
    for (int i = 0; i < iters; ++i) {
        long long idx = (long long)tid + (long long)i * nthreads;
        v4f x = __builtin_nontemporal_load(&av[idx]);   // global_load_b128 th:NT
        v4f y = __builtin_nontemporal_load(&bv[idx]);
        s0 += __builtin_fabsf(x[0] - y[0]);
        s1 += __builtin_fabsf(x[1] - y[1]);
        s2 += __builtin_fabsf(x[2] - y[2]);
        s3 += __builtin_fabsf(x[3] - y[3]);
    }

    // Generic tail (empty for N = 2^26, kept for size-safety).
    long long done = (long long)iters * nthreads * 4;
    for (long long e = done + tid; e < n; e += nthreads)
        s0 += __builtin_fabsf(a[e] - b[e]);

    // Wave reduction on the matrix pipe (EXEC is all-ones here).
    float wsum = wave_sum_wmma((s0 + s1) + (s2 + s3));

    // Cross-wave reduction through LDS (8 waves per 256-thread block).
    __shared__ float lds[8];
    const int lane = threadIdx.x & 31;
    const int wave = threadIdx.x >> 5;
    if (lane == 0) lds[wave] = wsum;
    __syncthreads();
    if (threadIdx.x == 0) {
        float s = 0.0f;
        const int nw = blockDim.x >> 5;
        for (int w = 0; w < nw; ++w) s += lds[w];   // deterministic order
        partials[blockIdx.x] = s;
    }
}

__global__ void __launch_bounds__(256)
l1_final_kernel(const float* __restrict__ partials, int nparts,
                float* __restrict__ out, float inv_rows) {
    float acc = 0.0f;
    // nparts is a multiple of blockDim.x -> uniform loop, EXEC stays all-ones.
    for (int i = threadIdx.x; i < nparts; i += blockDim.x)
        acc += partials[i];

    float wsum = wave_sum_wmma(acc);

    __shared__ float lds[8];
    const int lane = threadIdx.x & 31;
    const int wave = threadIdx.x >> 5;
    if (lane == 0) lds[wave] = wsum;
    __syncthreads();
    if (threadIdx.x == 0) {
        float s = 0.0f;
        const int nw = blockDim.x >> 5;
        for (int w = 0; w < nw; ++w) s += lds[w];
        out[0] = s * inv_rows;
    }
}

extern "C" void kernel_launch(void* const* d_in, const int* in_sizes, int n_in,
                              void* d_out, int out_size, void* d_ws, size_t ws_size,
                              hipStream_t stream) {
    const float* a   = (const float*)d_in[0];   // out
    const float* b   = (const float*)d_in[1];   // target
    float*       out = (float*)d_out;           // scalar f32
    float*       ws  = (float*)d_ws;            // block partials

    const long long n = (long long)in_sizes[0]; // 2^26 flat f32 elements
    const int BLOCK = 256;                      // 8 wave32s per block
    const int GRID  = 4096;                     // 2^20 lanes -> 16 b128 iters/lane

    const long long nvec  = n / 4;
    const int nthreads    = GRID * BLOCK;
    const int iters       = (int)(nvec / nthreads);

    l1_partial_kernel<<<GRID, BLOCK, 0, stream>>>(a, b, ws, n, iters);

    const long long rows = n / 8;               // N_FEAT = 8 per reference
    const float inv_rows = 1.0f / (float)rows;
    l1_final_kernel<<<1, BLOCK, 0, stream>>>(ws, GRID, out, inv_rows);
}